// Backflow_45586782879785
// MI455X (gfx1250) — compile-verified
//
#include <hip/hip_runtime.h>
#include <hip/hip_bf16.h>

typedef __attribute__((ext_vector_type(2))) float v2f;
typedef __attribute__((ext_vector_type(8))) float v8f;

#define B_SAMP 8192
#define NUM_O  64
#define NUM_E  16
#define NUM_D  16
#define NUM_H  4

// ---------------------------------------------------------------------------
// Kernel 1: h2 = relu(relu(x@W1+b1)@W2+b2)  and  sel = first 16 nonzero idx.
// Layer-1 GEMM (B x 64 @ 64 x 4) runs on V_WMMA_F32_16X16X4_F32:
// each wave owns a 16-sample tile, K=64 -> 16 chained WMMAs (N padded to 16).
// 128 threads = 4 waves = 64 samples per block.
// B-fragment lane masking is done arithmetically (in-bounds address * mask)
// so loads stay unconditional -- no exec-mask branching around the WMMAs.
// ---------------------------------------------------------------------------
__global__ __launch_bounds__(128) void bf_prep_kernel(
    const float* __restrict__ x,  const float* __restrict__ W1,
    const float* __restrict__ b1, const float* __restrict__ W2,
    const float* __restrict__ b2,
    float* __restrict__ h2_out, int* __restrict__ sel_out)
{
    __shared__ float h1s[4][16][16];

    const int t    = threadIdx.x;
    const int wave = t >> 5;        // 0..3
    const int lane = t & 31;
    const int hh   = lane >> 4;     // half-wave: 0 or 1
    const int m    = lane & 15;     // M index (sample-in-tile for A, N col for B)
    const int mc   = m & 3;         // clamped col for in-bounds W1/b1 reads
    const float nmask = (m < NUM_H) ? 1.0f : 0.0f;  // N-padding mask (cols 4..15 = 0)
    const int s0   = blockIdx.x * 64 + wave * 16;

    v8f acc = {};
    #pragma unroll
    for (int kk = 0; kk < 16; ++kk) {
        const int k0 = kk * 4 + hh * 2;   // A 16x4 f32 layout: lanes0-15 K=0,1; lanes16-31 K=2,3
        const float2 av =
            *reinterpret_cast<const float2*>(x + (size_t)(s0 + m) * NUM_O + k0);
        v2f a; a[0] = av.x; a[1] = av.y;
        // B 4x16: row k striped across lanes; cols >= 4 are zero padding.
        // Address is always in-bounds; padding applied via multiply.
        v2f b;
        b[0] = W1[k0 * NUM_H + mc]       * nmask;
        b[1] = W1[(k0 + 1) * NUM_H + mc] * nmask;
        acc = __builtin_amdgcn_wmma_f32_16x16x4_f32(
            false, a, false, b, (short)0, acc, false, false);
    }
    // C layout: VGPR r -> M = r + 8*hh, N = m.  Apply bias + relu, stash in LDS.
    const float bias1 = b1[mc] * nmask;
    #pragma unroll
    for (int r = 0; r < 8; ++r) {
        h1s[wave][r + 8 * hh][m] = fmaxf(acc[r] + bias1, 0.0f);
    }
    __syncthreads();

    // Phase 2: one thread per sample (64 samples/block) -> layer 2 + sel scan.
    if (t < 64) {
        const int s = blockIdx.x * 64 + t;
        float h1v[NUM_H];
        #pragma unroll
        for (int n = 0; n < NUM_H; ++n) h1v[n] = h1s[t >> 4][t & 15][n];
        #pragma unroll
        for (int n = 0; n < NUM_H; ++n) {
            float v = b2[n];
            #pragma unroll
            for (int mm = 0; mm < NUM_H; ++mm) v += h1v[mm] * W2[mm * NUM_H + n];
            h2_out[s * NUM_H + n] = fmaxf(v, 0.0f);
        }
        // sel: indices of first 16 nonzeros, fill 0 (matches jnp.nonzero size=16)
        #pragma unroll
        for (int i = 0; i < NUM_E; ++i) sel_out[s * NUM_E + i] = 0;
        int cnt = 0;
        for (int o = 0; o < NUM_O; ++o) {
            float xv = x[(size_t)s * NUM_O + o];
            if (xv != 0.0f && cnt < NUM_E) { sel_out[s * NUM_E + cnt] = o; ++cnt; }
        }
    }
}

// ---------------------------------------------------------------------------
// Kernel 2: one sample per 256-thread block. 16-lane group = det index d,
// lane r owns matrix row r. Build row from gathered Wg/bg (L2-resident),
// Gaussian elimination with __shfl(width=16) row broadcasts (lowers to VALU
// lane-permutes on gfx1250), det = prod(pivots).
// ---------------------------------------------------------------------------
__global__ __launch_bounds__(256) void bf_det_kernel(
    const float* __restrict__ Wg, const float* __restrict__ bg,
    const float* __restrict__ bfp,
    const float* __restrict__ h2, const int* __restrict__ sel,
    float* __restrict__ out)
{
    __shared__ float dets[NUM_D];

    const int s = blockIdx.x;
    const int t = threadIdx.x;
    const int d = t >> 4;    // det index 0..15
    const int r = t & 15;    // row index 0..15

    float hk[NUM_H];
    #pragma unroll
    for (int k = 0; k < NUM_H; ++k) hk[k] = h2[s * NUM_H + k];
    const int sr = sel[s * NUM_E + r];

    // row[j] = bg[d, sr, j] + sum_k hk[k] * Wg[k, d, sr, j]
    float row[16];
    {
        const float4* bgp =
            reinterpret_cast<const float4*>(bg + ((size_t)d * NUM_O + sr) * NUM_E);
        #pragma unroll
        for (int q = 0; q < 4; ++q) {
            float4 v = bgp[q];
            row[4*q+0] = v.x; row[4*q+1] = v.y; row[4*q+2] = v.z; row[4*q+3] = v.w;
        }
        #pragma unroll
        for (int k = 0; k < NUM_H; ++k) {
            const float4* wp = reinterpret_cast<const float4*>(
                Wg + (((size_t)k * NUM_D + d) * NUM_O + sr) * NUM_E);
            #pragma unroll
            for (int q = 0; q < 4; ++q) {
                float4 v = wp[q];
                row[4*q+0] += hk[k] * v.x; row[4*q+1] += hk[k] * v.y;
                row[4*q+2] += hk[k] * v.z; row[4*q+3] += hk[k] * v.w;
            }
        }
    }

    // Gaussian elimination (no pivoting; dense random matrices). All lanes of
    // the half-wave execute every shuffle (ternaries select, no divergence).
    float det = 1.0f;
    #pragma unroll
    for (int k = 0; k < 16; ++k) {
        const float piv = __shfl(row[k], k, 16);
        det *= piv;
        const float f = (r > k && piv != 0.0f) ? (row[k] / piv) : 0.0f;
        #pragma unroll
        for (int j = k + 1; j < 16; ++j) {
            row[j] -= f * __shfl(row[j], k, 16);
        }
    }

    if (r == 0) dets[d] = det;
    __syncthreads();

    if (t == 0) {
        float accum = 0.0f;
        #pragma unroll
        for (int dd = 0; dd < NUM_D; ++dd) accum += dets[dd] * bfp[dd];
        int ssum = 0;
        #pragma unroll
        for (int i = 0; i < NUM_E; ++i) ssum += sel[s * NUM_E + i];
        out[s] = (ssum == 0) ? 0.0f : accum;
    }
}

extern "C" void kernel_launch(void* const* d_in, const int* in_sizes, int n_in,
                              void* d_out, int out_size, void* d_ws, size_t ws_size,
                              hipStream_t stream) {
    const float* x   = (const float*)d_in[0];
    const float* W1  = (const float*)d_in[1];
    const float* b1  = (const float*)d_in[2];
    const float* W2  = (const float*)d_in[3];
    const float* b2  = (const float*)d_in[4];
    const float* Wg  = (const float*)d_in[5];
    const float* bg  = (const float*)d_in[6];
    const float* bfp = (const float*)d_in[7];
    float* out = (float*)d_out;

    float* h2  = (float*)d_ws;                                    // B*4 floats
    int*   sel = (int*)((char*)d_ws + (size_t)B_SAMP * NUM_H * sizeof(float)); // B*16 ints

    bf_prep_kernel<<<B_SAMP / 64, 128, 0, stream>>>(x, W1, b1, W2, b2, h2, sel);
    bf_det_kernel<<<B_SAMP, 256, 0, stream>>>(Wg, bg, bfp, h2, sel, out);
}